// GIN_10419590660822
// MI455X (gfx1250) — compile-verified
//
#include <hip/hip_runtime.h>
#include <hip/hip_bf16.h>

#define N_NODES 100000
#define N_EDGES 600000
#define DIM     128
#define DOUT    40
#define LDA     132   // padded LDS stride (floats): conflict-free WMMA fragment reads

typedef __attribute__((ext_vector_type(2))) float v2f;
typedef __attribute__((ext_vector_type(8))) float v8f;
typedef __attribute__((ext_vector_type(4))) int   v4i;

#define AS_GLOBAL __attribute__((address_space(1)))
#define AS_LDS    __attribute__((address_space(3)))

#if __has_builtin(__builtin_amdgcn_global_load_async_to_lds_b128) && \
    __has_builtin(__builtin_amdgcn_s_wait_asynccnt)
#define HAVE_ASYNC_LDS 1
#else
#define HAVE_ASYNC_LDS 0
#endif

// C[i] = (1 + eps[layer]) * x[i], float4-vectorized
__global__ __launch_bounds__(256) void gin_scale(const float* __restrict__ x,
                                                 const float* __restrict__ eps, int layer,
                                                 float* __restrict__ c, int n4) {
    float s = 1.0f + eps[layer];
    int i = blockIdx.x * blockDim.x + threadIdx.x;
    if (i < n4) {
        float4 v = ((const float4*)x)[i];
        ((float4*)c)[i] = make_float4(v.x * s, v.y * s, v.z * s, v.w * s);
    }
}

// one wave per edge: lane l adds x[src][4l..4l+3] into c[dst][4l..4l+3]
__global__ __launch_bounds__(256) void gin_scatter(const float* __restrict__ x,
                                                   const int* __restrict__ ei,
                                                   float* __restrict__ c, int n_edges) {
    int wave = (blockIdx.x * blockDim.x + threadIdx.x) >> 5;
    int lane = threadIdx.x & 31;
    if (wave >= n_edges) return;
    int s = ei[wave];
    int d = ei[n_edges + wave];
    float4 v = *(const float4*)(x + (size_t)s * DIM + 4 * lane);
    float* cd = c + (size_t)d * DIM + 4 * lane;
    atomicAdd(cd + 0, v.x);
    atomicAdd(cd + 1, v.y);
    atomicAdd(cd + 2, v.z);
    atomicAdd(cd + 3, v.w);
}

// stage one 16-byte chunk of the A tile into LDS (async DMA if available)
__device__ __forceinline__ void stage_a_chunk(const float* __restrict__ A,
                                              float* lds_a, int m, int k,
                                              int node, int n_nodes) {
#if HAVE_ASYNC_LDS
    if (node < n_nodes) {
        __builtin_amdgcn_global_load_async_to_lds_b128(
            (AS_GLOBAL v4i*)(A + (size_t)node * DIM + k),
            (AS_LDS v4i*)&lds_a[m * LDA + k], 0, 0);
    } else {
        *(float4*)&lds_a[m * LDA + k] = make_float4(0.f, 0.f, 0.f, 0.f);
    }
#else
    float4 av = make_float4(0.f, 0.f, 0.f, 0.f);
    if (node < n_nodes) av = *(const float4*)(A + (size_t)node * DIM + k);
    *(float4*)&lds_a[m * LDA + k] = av;
#endif
}

__device__ __forceinline__ void stage_a_fence() {
#if HAVE_ASYNC_LDS
    __builtin_amdgcn_s_wait_asynccnt(0);   // my DMA writes landed in LDS
#endif
    __syncthreads();                        // publish to all waves in the block
}

// Y[64-node tile] = A(64x128, in LDS) @ W(128x128, global/L2) + bias
// 8 waves: wave w owns 16 output cols, iterates 4 node subtiles; fp32 WMMA 16x16x4
__global__ __launch_bounds__(256) void gin_gemm128(const float* __restrict__ A,
                                                   const float* __restrict__ W,
                                                   const float* __restrict__ bias,
                                                   float* __restrict__ Y, int n_nodes) {
    __shared__ float lds_a[64 * LDA];
    int tid  = threadIdx.x;
    int base = blockIdx.x * 64;

    // stage combined-feature tile: 64 x 128, coalesced 16B chunks, padded rows
    #pragma unroll
    for (int it = 0; it < 8; ++it) {
        int idx = (it * 256 + tid) * 4;          // 8192 floats total
        int m = idx >> 7, k = idx & 127;
        stage_a_chunk(A, lds_a, m, k, base + m, n_nodes);
    }
    stage_a_fence();

    int wv = tid >> 5;
    int lane = tid & 31;
    int g = lane >> 4;        // half-wave group
    int n = lane & 15;        // col within tile / row within A fragment
    int c0 = wv * 16;

    v8f acc[4] = {v8f{}, v8f{}, v8f{}, v8f{}};
    for (int k0 = 0; k0 < DIM; k0 += 4) {
        int kk = k0 + 2 * g;                     // A/B fragment: K = 2g + v
        v2f b;
        b.x = W[(size_t)kk * DIM + c0 + n];      // B[k][n] striped over lanes
        b.y = W[(size_t)(kk + 1) * DIM + c0 + n];
        #pragma unroll
        for (int mt = 0; mt < 4; ++mt) {
            v2f a = *(const v2f*)&lds_a[(mt * 16 + n) * LDA + kk];  // A[m=n][k]
            acc[mt] = __builtin_amdgcn_wmma_f32_16x16x4_f32(
                false, a, false, b, (short)0, acc[mt], false, false);
        }
    }

    float bv = bias[c0 + n];
    #pragma unroll
    for (int mt = 0; mt < 4; ++mt) {
        #pragma unroll
        for (int r = 0; r < 8; ++r) {
            int node = base + mt * 16 + 8 * g + r;   // D row = r + 8g
            if (node < n_nodes)
                Y[(size_t)node * DIM + c0 + n] = acc[mt][r] + bv;
        }
    }
}

// out[64-node tile] = A(64x128) @ W_head(128x40) + b_head; cols padded to 48
__global__ __launch_bounds__(256) void gin_head(const float* __restrict__ A,
                                                const float* __restrict__ Wh,
                                                const float* __restrict__ bh,
                                                float* __restrict__ out, int n_nodes) {
    __shared__ float lds_a[64 * LDA];
    __shared__ float lds_w[48 * LDA];            // W_head^T, zero-padded cols 40..47
    int tid  = threadIdx.x;
    int base = blockIdx.x * 64;

    #pragma unroll
    for (int it = 0; it < 8; ++it) {
        int idx = (it * 256 + tid) * 4;
        int m = idx >> 7, k = idx & 127;
        stage_a_chunk(A, lds_a, m, k, base + m, n_nodes);
    }
    for (int idx = tid; idx < 48 * DIM; idx += 256) {
        int cc = idx >> 7, k = idx & 127;
        lds_w[cc * LDA + k] = (cc < DOUT) ? Wh[(size_t)k * DOUT + cc] : 0.0f;
    }
    stage_a_fence();

    int wv = tid >> 5;
    int lane = tid & 31;
    int g = lane >> 4;
    int n = lane & 15;

    for (int t = wv; t < 12; t += 8) {           // 4 M-subtiles x 3 col-tiles
        int mt = t / 3, nt = t % 3;
        int c0 = nt * 16;
        v8f acc = {};
        for (int k0 = 0; k0 < DIM; k0 += 4) {
            int kk = k0 + 2 * g;
            v2f b = *(const v2f*)&lds_w[(c0 + n) * LDA + kk];
            v2f a = *(const v2f*)&lds_a[(mt * 16 + n) * LDA + kk];
            acc = __builtin_amdgcn_wmma_f32_16x16x4_f32(
                false, a, false, b, (short)0, acc, false, false);
        }
        int col = c0 + n;
        float bv = (col < DOUT) ? bh[col] : 0.0f;
        #pragma unroll
        for (int r = 0; r < 8; ++r) {
            int node = base + mt * 16 + 8 * g + r;
            if (node < n_nodes && col < DOUT)
                out[(size_t)node * DOUT + col] = acc[r] + bv;
        }
    }
}

extern "C" void kernel_launch(void* const* d_in, const int* in_sizes, int n_in,
                              void* d_out, int out_size, void* d_ws, size_t ws_size,
                              hipStream_t stream) {
    const float* x   = (const float*)d_in[0];
    const int*   ei  = (const int*)d_in[1];     // edge_index[2][N_EDGES]
    const float* eps = (const float*)d_in[2];   // [3]
    const float* Ws  = (const float*)d_in[3];   // [3][128][128]
    const float* bs  = (const float*)d_in[4];   // [3][128]
    const float* Wh  = (const float*)d_in[5];   // [128][40]
    const float* bh  = (const float*)d_in[6];   // [40]
    float* out = (float*)d_out;

    float* C = (float*)d_ws;                    // combine buffer, 100000*128 f32
    float* Y = C + (size_t)N_NODES * DIM;       // layer output,  100000*128 f32

    const int n4 = N_NODES * DIM / 4;           // 3,200,000 float4s
    dim3 thr(256);
    dim3 grid_scale((n4 + 255) / 256);
    dim3 grid_scatter(N_EDGES / 8);             // 8 waves (edges) per block
    dim3 grid_gemm((N_NODES + 63) / 64);

    for (int l = 0; l < 3; ++l) {
        const float* xin = (l == 0) ? x : Y;
        gin_scale  <<<grid_scale,   thr, 0, stream>>>(xin, eps, l, C, n4);
        gin_scatter<<<grid_scatter, thr, 0, stream>>>(xin, ei, C, N_EDGES);
        gin_gemm128<<<grid_gemm,    thr, 0, stream>>>(C, Ws + (size_t)l * DIM * DIM,
                                                      bs + (size_t)l * DIM, Y, N_NODES);
    }
    gin_head<<<grid_gemm, thr, 0, stream>>>(Y, Wh, bh, out, N_NODES);
}